// InverseDiscreteWaveletTransform2D_88734024335380
// MI455X (gfx1250) — compile-verified
//
#include <hip/hip_runtime.h>

// Inverse Haar DWT (B,C,256,256)x4 -> (B,C,512,512), fp32.
// Bandwidth-bound: 1 GiB total traffic, ~46us floor at 23.3 TB/s.
// Strategy: express the 4x4 Haar synthesis as V_WMMA_F32_16X16X4_F32 (K=4,
// full fp32 -> bit-accurate class of arithmetic). One wave owns one half-res
// row; each WMMA produces 16 half-res pixels (64 output floats) with only
// 2 dense 128B loads + 2 dense 128B stores. Non-temporal hints since every
// byte is touched exactly once and the stream >> 192MB L2.

typedef float v2f __attribute__((ext_vector_type(2)));
typedef float v8f __attribute__((ext_vector_type(8)));

__global__ __launch_bounds__(256) void ihaar_wmma_kernel(
    const float* __restrict__ ll, const float* __restrict__ lh,
    const float* __restrict__ hl, const float* __restrict__ hh,
    float* __restrict__ out, int nrows)
{
    const int W2 = 256;   // half-res width
    const int OW = 512;   // output width

    const int lane = threadIdx.x & 31;
    const int wave = blockIdx.x * (blockDim.x >> 5) + (threadIdx.x >> 5);
    if (wave >= nrows) return;                 // wave-uniform: EXEC stays all-1s

    const int m    = lane & 15;                // position-in-tile / A row
    const int half = lane >> 4;                // K half-split across half-waves

    // ---- A matrix (16x4 fp32 Haar synthesis, constant) ----
    // M rows: 0-3 -> combo a, 4-7 -> c, 8-11 -> b, 12-15 -> d
    // K: 0=ll, 1=lh, 2=hl, 3=hh.  Lane layout: a.x=A[m][2*half], a.y=A[m][2*half+1]
    const int co   = m >> 2;                   // 0=a,1=c,2=b,3=d
    const float s_ll = 0.5f;
    const float s_lh = (co & 1)              ?  0.5f : -0.5f;  // + for c,d
    const float s_hl = (co >= 2)             ?  0.5f : -0.5f;  // + for b,d
    const float s_hh = (co == 0 || co == 3)  ?  0.5f : -0.5f;  // + for a,d
    v2f av;
    av.x = half ? s_hl : s_ll;
    av.y = half ? s_hh : s_lh;

    // ---- per-lane streaming pointers ----
    // B-VGPR0 sources ll (lanes 0-15) / hl (lanes 16-31); B-VGPR1: lh / hh.
    const long in_off = (long)wave * W2 + m;
    const float* __restrict__ pB0 = (half ? hl : ll) + in_off;
    const float* __restrict__ pB1 = (half ? hh : lh) + in_off;

    const long bc = (long)(wave >> 8);         // b*C + c
    const long i  = (long)(wave & 255);        // half-res row
    // lane L -> output column 2*m + half: lanes 0-15 even cols (a/c),
    // lanes 16-31 odd cols (b/d); 32 lanes cover 128 contiguous bytes.
    float* __restrict__ outT = out + bc * (long)(OW * OW) + i * (long)(2 * OW)
                                   + (2 * m + half);
    float* __restrict__ outB = outT + OW;

    const v8f cz = {};

#pragma unroll
    for (int j0 = 0; j0 < W2; j0 += 16) {
        v2f bv;
        bv.x = __builtin_nontemporal_load(pB0 + j0);
        bv.y = __builtin_nontemporal_load(pB1 + j0);
        v8f d = __builtin_amdgcn_wmma_f32_16x16x4_f32(
            /*neg_a=*/false, av, /*neg_b=*/false, bv,
            /*c_mod=*/(short)0, cz, /*reuse_a=*/false, /*reuse_b=*/false);
        // D VGPR0 = rows (a | b) -> top output row; VGPR4 = (c | d) -> bottom.
        __builtin_nontemporal_store(d[0], outT + 2 * j0);
        __builtin_nontemporal_store(d[4], outB + 2 * j0);
    }
}

extern "C" void kernel_launch(void* const* d_in, const int* in_sizes, int n_in,
                              void* d_out, int out_size, void* d_ws, size_t ws_size,
                              hipStream_t stream) {
    const float* ll = (const float*)d_in[0];
    const float* lh = (const float*)d_in[1];
    const float* hl = (const float*)d_in[2];
    const float* hh = (const float*)d_in[3];
    float* out = (float*)d_out;

    const int W2    = 256;
    const int nrows = in_sizes[0] / W2;        // B*C*H2 = 131072
    const int waves_per_block = 8;             // 256 threads, wave32
    const int nblocks = (nrows + waves_per_block - 1) / waves_per_block;

    ihaar_wmma_kernel<<<nblocks, 256, 0, stream>>>(ll, lh, hl, hh, out, nrows);
}